// LearnablePeakExtractor_17987323035999
// MI455X (gfx1250) — compile-verified
//
#include <hip/hip_runtime.h>
#include <stdint.h>

// ---------------------------------------------------------------------------
// LearnablePeakExtractor for MI455X (gfx1250)
//
// peak_map: [256, 131072] f32, logit_thresh: scalar f32
//   thresh     = sigmoid(logit_thresh)
//   gate       = sigmoid(10*(x - thresh))
//   pooled     = sliding max, window 5, replicate pad (== index clamp)
//   local_mask = sigmoid(10*(x - pooled))
//   smooth     = x * gate * local_mask
//   mask       = smooth >= thresh          (stored as 0.0f / 1.0f)
//
// HBM-bound (~402 MB traffic -> ~17us @ 23.3 TB/s). Strategy: stage each
// 2048-element row tile into LDS with CDNA5 async global->LDS B128 DMA
// (ASYNCcnt), compute window-max from LDS with aligned float4 reads, stream
// results out with non-temporal stores so the write-once outputs don't
// thrash the 192 MB L2.
// ---------------------------------------------------------------------------

#define L_LEN   131072
#define ROWS    256
#define TILE    2048          // elements per block
#define TPB     256           // threads per block (8 wave32s)
#define SHARP   10.0f

typedef float v4f __attribute__((ext_vector_type(4)));

__device__ __forceinline__ float fsig(float z) {
    return 1.0f / (1.0f + __expf(-z));
}

__device__ __forceinline__ float max5(float a, float b, float c, float d, float e) {
    return fmaxf(fmaxf(fmaxf(a, b), fmaxf(c, d)), e);
}

__global__ __launch_bounds__(TPB)
void peak_extract_kernel(const float* __restrict__ peak,
                         const float* __restrict__ logit,
                         float* __restrict__ out_sp,
                         float* __restrict__ out_mk) {
    // LDS layout: tile[2..3] = left halo, tile[4 .. TILE+3] = data,
    // tile[TILE+4 .. TILE+5] = right halo. Data element j -> tile[4 + j],
    // so async B128 destinations (j multiple of 4) are 16B aligned.
    __shared__ __align__(16) float tile[TILE + 8];

    const int    tid     = threadIdx.x;
    const int    row     = blockIdx.y;
    const int    t0      = blockIdx.x * TILE;
    const size_t rowbase = (size_t)row * (size_t)L_LEN;

    // LDS byte offset of tile[] (low 32 bits of the flat address are the
    // wave-local LDS byte offset on gfx1250).
    const uint32_t lds_base = (uint32_t)(size_t)(&tile[0]);
    // Global byte offset of this tile from the peak_map base (fits in u32:
    // 256*131072*4 = 128 MiB).
    const uint32_t gbyte = (uint32_t)((rowbase + (size_t)t0) * sizeof(float));

    // --- async DMA: main tile, two float4 quads per thread -----------------
#pragma unroll
    for (int k = 0; k < 2; ++k) {
        const uint32_t q       = (uint32_t)tid + (uint32_t)k * TPB; // quad index
        const uint32_t lds_off = lds_base + 16u + q * 16u;          // tile[4 + 4q]
        const uint32_t g_off   = gbyte + q * 16u;
        // GVS mode: mem = SADDR(64) + VADDR(u32) ; LDS dst = VDST byte offset.
        asm volatile("global_load_async_to_lds_b128 %0, %1, %2"
                     :: "v"(lds_off), "v"(g_off), "s"(peak)
                     : "memory");
    }

    // --- halo: 4 scalars, branchless, row-edge clamp (replicate padding) ---
    if (tid < 4) {
        // tid 0 -> (t0-2,      tile[2])        tid 1 -> (t0-1,      tile[3])
        // tid 2 -> (t0+TILE,   tile[TILE+4])   tid 3 -> (t0+TILE+1, tile[TILE+5])
        const int side = (tid >> 1) * (TILE + 2);
        const int dst  = 2 + (tid & 1) + side;
        int pos        = t0 - 2 + (tid & 1) + side;
        pos = pos < 0 ? 0 : (pos > L_LEN - 1 ? L_LEN - 1 : pos);
        tile[dst] = peak[rowbase + (size_t)pos];
    }

    // Wait for this wave's async DMA to land in LDS, then sync the workgroup
    // (every wave has drained its own ASYNCcnt before signaling the barrier,
    // so after the barrier all tile data is visible to all waves).
    asm volatile("s_wait_asynccnt 0" ::: "memory");
    __syncthreads();

    const float thresh = fsig(logit[0]);

    const v4f* vt = (const v4f*)tile;

#pragma unroll
    for (int k = 0; k < 2; ++k) {
        const int q = tid + k * TPB;
        const int c = q * 4;                 // local element index of quad
        // Window for data j=c..c+3 spans tile[c+2 .. c+9]; read 3 aligned v4s.
        const v4f A = vt[q];                 // tile[c   .. c+3]
        const v4f B = vt[q + 1];             // tile[c+4 .. c+7]  (the data)
        const v4f C = vt[q + 2];             // tile[c+8 .. c+11] (.z/.w unused)

        const float x0 = B.x, x1 = B.y, x2 = B.z, x3 = B.w;
        const float p0 = max5(A.z, A.w, B.x, B.y, B.z);
        const float p1 = max5(A.w, B.x, B.y, B.z, B.w);
        const float p2 = max5(B.x, B.y, B.z, B.w, C.x);
        const float p3 = max5(B.y, B.z, B.w, C.x, C.y);

        v4f sp, mk;
        sp.x = x0 * fsig(SHARP * (x0 - thresh)) * fsig(SHARP * (x0 - p0));
        sp.y = x1 * fsig(SHARP * (x1 - thresh)) * fsig(SHARP * (x1 - p1));
        sp.z = x2 * fsig(SHARP * (x2 - thresh)) * fsig(SHARP * (x2 - p2));
        sp.w = x3 * fsig(SHARP * (x3 - thresh)) * fsig(SHARP * (x3 - p3));

        mk.x = sp.x >= thresh ? 1.0f : 0.0f;
        mk.y = sp.y >= thresh ? 1.0f : 0.0f;
        mk.z = sp.z >= thresh ? 1.0f : 0.0f;
        mk.w = sp.w >= thresh ? 1.0f : 0.0f;

        const size_t o = rowbase + (size_t)t0 + (size_t)c;
        __builtin_nontemporal_store(sp, (v4f*)(out_sp + o));
        __builtin_nontemporal_store(mk, (v4f*)(out_mk + o));
    }
}

extern "C" void kernel_launch(void* const* d_in, const int* in_sizes, int n_in,
                              void* d_out, int out_size, void* d_ws, size_t ws_size,
                              hipStream_t stream) {
    (void)in_sizes; (void)n_in; (void)out_size; (void)d_ws; (void)ws_size;

    const float* peak  = (const float*)d_in[0];  // [256, 131072] f32
    const float* logit = (const float*)d_in[1];  // scalar f32

    float* out_sp = (float*)d_out;                              // smooth_peaks
    float* out_mk = out_sp + (size_t)ROWS * (size_t)L_LEN;      // mask (0/1 f32)

    dim3 grid(L_LEN / TILE, ROWS);   // 64 x 256 = 16384 blocks
    dim3 block(TPB);
    peak_extract_kernel<<<grid, block, 0, stream>>>(peak, logit, out_sp, out_mk);
}